// TopoFilter_62294205662067
// MI455X (gfx1250) — compile-verified
//
#include <hip/hip_runtime.h>
#include <math.h>

// CDNA5 / gfx1250: wave32, WMMA fp32 16x16x4, 320KB LDS per WGP.

typedef float v2f __attribute__((ext_vector_type(2)));
typedef float v8f __attribute__((ext_vector_type(8)));

#define N_TOT   8192
#define DIM     512
#define DHALF   256          // float2 per row
#define ROWS_PB 128          // rows per workgroup (8 waves x 16)
#define STRIDE  514          // LDS row stride in floats (512 + 2 pad: bank-conflict-free b64 frag loads)
#define KP1     6            // keep 6 smallest (self + 5 NN)

__device__ __forceinline__ void insert6(float t[KP1], float v) {
  // t sorted ascending; keep 6 smallest seen.
  if (v < t[KP1 - 1]) {
    t[KP1 - 1] = v;
#pragma unroll
    for (int s = KP1 - 1; s > 0; --s) {
      float a = t[s - 1], b = t[s];
      t[s - 1] = fminf(a, b);
      t[s]     = fmaxf(a, b);
    }
  }
}

// One wave per row: sqn[row] = sum_d x[row][d]^2
__global__ void __launch_bounds__(256)
sqnorm_kernel(const float* __restrict__ x, float* __restrict__ sqn) {
  int wave = threadIdx.x >> 5;
  int lane = threadIdx.x & 31;
  int row  = (blockIdx.x << 3) + wave;
  const float* p = x + (size_t)row * DIM;
  float s = 0.f;
#pragma unroll
  for (int d = 0; d < DIM; d += 128) {
    float4 v = *(const float4*)(p + d + (lane << 2));
    s += v.x * v.x + v.y * v.y + v.z * v.z + v.w * v.w;
  }
#pragma unroll
  for (int m = 16; m >= 1; m >>= 1) s += __shfl_xor(s, m, 32);
  if (lane == 0) sqn[row] = s;
}

__global__ void __launch_bounds__(256)
topo_kernel(const float* __restrict__ X, const float* __restrict__ sqn,
            float* __restrict__ out) {
  extern __shared__ float smem[];
  float* smA = smem;                      // 128 x 514 floats (this block's rows)
  float* smB = smA + ROWS_PB * STRIDE;    // 16 x 514 floats (streamed column tile)
  float* smW = smB + 16 * STRIDE;         // 128 row weights

  const int  tid     = threadIdx.x;
  const int  wave    = tid >> 5;
  const int  lane    = tid & 31;
  const int  lcls    = lane & 15;         // column class / fragment row index
  const bool hi      = lane >= 16;
  const int  iBase   = blockIdx.x * ROWS_PB;
  const int  waveRow = iBase + wave * 16;

  // Stage the 128 A-rows into LDS once (padded stride).
  for (int idx = tid; idx < ROWS_PB * DHALF; idx += 256) {
    int r = idx >> 8, c = idx & 255;
    *(float2*)&smA[r * STRIDE + (c << 1)] =
        *(const float2*)&X[(size_t)(iBase + r) * DIM + (c << 1)];
  }

  // Row norms for this wave's 8 rows (C-matrix: lanes<16 -> rows 0..7, lanes>=16 -> rows 8..15).
  float sqi[8];
#pragma unroll
  for (int r = 0; r < 8; ++r)
    sqi[r] = sqn[waveRow + (hi ? 8 : 0) + r];

  // Per-lane streaming top-6 of SQUARED distances (sqrt is monotone -> same
  // selection; sqrt applied only to the 5 winners per row at the end).
  float t[8][KP1];
#pragma unroll
  for (int r = 0; r < 8; ++r)
#pragma unroll
    for (int s = 0; s < KP1; ++s) t[r][s] = 3.0e38f;

  // WMMA f32 16x16x4 fragment addressing:
  // lanes 0-15 hold K{k,k+1}, lanes 16-31 hold K{k+2,k+3}; M/N index = lane%16.
  const int aOff = (wave * 16 + lcls) * STRIDE + (hi ? 2 : 0);
  const int bOff = lcls * STRIDE + (hi ? 2 : 0);

  for (int jBase = 0; jBase < N_TOT; jBase += 16) {
    __syncthreads();  // previous tile fully consumed
    for (int idx = tid; idx < 16 * DHALF; idx += 256) {
      int c = idx >> 8, d = idx & 255;
      *(float2*)&smB[c * STRIDE + (d << 1)] =
          *(const float2*)&X[(size_t)(jBase + c) * DIM + (d << 1)];
    }
    __syncthreads();

    v8f acc = {0.f, 0.f, 0.f, 0.f, 0.f, 0.f, 0.f, 0.f};
#pragma unroll 8
    for (int k = 0; k < DIM; k += 4) {
      v2f a = *(const v2f*)&smA[aOff + k];   // ds_load_b64: A frag {K,K+1}
      v2f b = *(const v2f*)&smB[bOff + k];   // ds_load_b64: B frag {K,K+1}
      acc = __builtin_amdgcn_wmma_f32_16x16x4_f32(
          /*neg_a=*/false, a, /*neg_b=*/false, b,
          /*c_mod=*/(short)0, acc, /*reuse_a=*/false, /*reuse_b=*/false);
    }

    float sqj = sqn[jBase + lcls];
#pragma unroll
    for (int r = 0; r < 8; ++r) {
      float d2 = sqi[r] + sqj - 2.0f * acc[r];   // squared distance, no sqrt here
      insert6(t[r], fmaxf(d2, 0.f));
    }
  }

  // Butterfly-merge the 16 column-class lists (xor 1,2,4,8 stays inside each
  // 16-lane half, matching the C-layout row split).
#pragma unroll
  for (int r = 0; r < 8; ++r) {
#pragma unroll
    for (int m = 1; m <= 8; m <<= 1) {
      float o[KP1];
#pragma unroll
      for (int s = 0; s < KP1; ++s) o[s] = __shfl_xor(t[r][s], m, 32);
#pragma unroll
      for (int s = 0; s < KP1; ++s) insert6(t[r], o[s]);
    }
    // t[0] is the squared self distance (global minimum ~0) -> drop it.
    // 5 raw v_sqrt_f32 per row, once -- no refinement sequence in any loop.
    float score = 0.2f * (__builtin_amdgcn_sqrtf(t[r][1]) +
                          __builtin_amdgcn_sqrtf(t[r][2]) +
                          __builtin_amdgcn_sqrtf(t[r][3]) +
                          __builtin_amdgcn_sqrtf(t[r][4]) +
                          __builtin_amdgcn_sqrtf(t[r][5]));
    float w = (score > 0.1f) ? __expf(-score) : 0.f;
    if (lcls == r) smW[wave * 16 + (hi ? 8 : 0) + r] = w;  // lanes r and 16+r write
  }

  // Scale this wave's 16 rows (rows already resident in LDS) and store.
  const float* aw = smA + wave * 16 * STRIDE;
  float* ow = out + (size_t)waveRow * DIM;
  for (int idx = lane; idx < 16 * DHALF; idx += 32) {
    int r = idx >> 8, c = idx & 255;
    float w = smW[wave * 16 + r];
    float2 xv = *(const float2*)&aw[r * STRIDE + (c << 1)];
    float2 ov = make_float2(xv.x * w, xv.y * w);
    *(float2*)&ow[((size_t)r << 9) + (c << 1)] = ov;
  }
}

extern "C" void kernel_launch(void* const* d_in, const int* in_sizes, int n_in,
                              void* d_out, int out_size, void* d_ws, size_t ws_size,
                              hipStream_t stream) {
  (void)in_sizes; (void)n_in;
  const float* X   = (const float*)d_in[0];
  float*       out = (float*)d_out;

  // 8192 fp32 row norms live in workspace (fallback: tail of d_out, which is
  // only overwritten by the final block's store phase).
  float* sqn = (ws_size >= (size_t)N_TOT * sizeof(float))
                   ? (float*)d_ws
                   : (out + (out_size - N_TOT));

  sqnorm_kernel<<<N_TOT / 8, 256, 0, stream>>>(X, sqn);

  const size_t smemBytes =
      (size_t)(ROWS_PB * STRIDE + 16 * STRIDE + ROWS_PB) * sizeof(float);  // ~290 KB <= 320 KB
  topo_kernel<<<N_TOT / ROWS_PB, 256, smemBytes, stream>>>(X, sqn, out);
}